// GAT_67173288509837
// MI455X (gfx1250) — compile-verified
//
#include <hip/hip_runtime.h>
#include <hip/hip_bf16.h>

typedef __attribute__((ext_vector_type(2))) float v2f;
typedef __attribute__((ext_vector_type(8))) float v8f;

#define F_IN 512
#define HID 8
#define HEADS 8
#define D1 64              // HEADS*HID
#define NCLS 16
#define NEG_SLOPE 0.2f
#define EPS_GAT 1e-16f

// ---- monotonic float<->uint encoding for atomic max on f32 via u32 ----
__device__ __forceinline__ unsigned fenc(float f) {
  unsigned u = __float_as_uint(f);
  return (u & 0x80000000u) ? ~u : (u | 0x80000000u);
}
__device__ __forceinline__ float fdec(unsigned e) {
  return (e & 0x80000000u) ? __uint_as_float(e ^ 0x80000000u) : __uint_as_float(~e);
}
#define ENC_NEG_INF 0x007FFFFFu   // fenc(-inf)

__device__ __forceinline__ float lrelu(float x) { return x > 0.f ? x : x * NEG_SLOPE; }

// ============================ fill kernels ============================
__global__ void fill_f32(float* p, float v, int n) {
  int i = blockIdx.x * blockDim.x + threadIdx.x;
  if (i < n) p[i] = v;
}
__global__ void fill_u32(unsigned* p, unsigned v, int n) {
  int i = blockIdx.x * blockDim.x + threadIdx.x;
  if (i < n) p[i] = v;
}

// ============================ GEMM1: xl1 = x @ W1  [N,512]x[512,64] ====
// block = 128 threads (4 waves). Each block: 16 rows. Each wave: one 16-col tile.
__global__ __launch_bounds__(128) void gemm1_kernel(const float* __restrict__ x,
                                                    const float* __restrict__ W1,
                                                    float* __restrict__ xl1, int nrows) {
  __shared__ float At[16 * 516];                 // padded stride 516 -> conflict-free
  const int tid = threadIdx.x;
  const int rowBase = blockIdx.x * 16;
  // cooperative A-tile load: 16 rows x 512 f32 = 2048 float4
  for (int i = tid; i < 2048; i += 128) {
    int r = i >> 7, cq = i & 127;
    int row = rowBase + r; if (row >= nrows) row = nrows - 1;
    *(float4*)&At[r * 516 + cq * 4] = ((const float4*)(x + (size_t)row * F_IN))[cq];
  }
  __syncthreads();

  const int wave = tid >> 5;          // 0..3 -> column tile
  const int lane = tid & 31;
  const int ln = lane & 15;           // M (A) / N (B,C) within 16
  const int khalf = (lane >> 4) << 1; // 0 or 2
  const int colBase = wave * 16;

  v8f c = {};
  #pragma unroll 4
  for (int k = 0; k < F_IN; k += 4) {
    v2f a = *(const v2f*)&At[ln * 516 + k + khalf];
    const float* wp = W1 + (size_t)(k + khalf) * D1 + colBase + ln;
    v2f b; b.x = wp[0]; b.y = wp[D1];
    c = __builtin_amdgcn_wmma_f32_16x16x4_f32(false, a, false, b, (short)0, c,
                                              false, false);
  }
  // C layout: VGPR i -> M = i + 8*(lane>=16), N = lane&15
  const int mBase = (lane >> 4) * 8;
  float* outp = xl1 + (size_t)rowBase * D1 + colBase;
  #pragma unroll
  for (int i = 0; i < 8; i++) {
    int row = rowBase + mBase + i;
    if (row < nrows) outp[(size_t)(mBase + i) * D1 + ln] = c[i];
  }
}

// ============================ attn1: per (node, head) dot products =====
__global__ void attn1_kernel(const float* __restrict__ xl1,
                             const float* __restrict__ att_src,
                             const float* __restrict__ att_dst,
                             float* __restrict__ a_src, float* __restrict__ a_dst, int n) {
  int t = blockIdx.x * blockDim.x + threadIdx.x;
  if (t >= n * HEADS) return;
  int node = t >> 3, hh = t & 7;
  const float* xp = xl1 + (size_t)node * D1 + hh * HID;
  const float* as = att_src + hh * HID;
  const float* ad = att_dst + hh * HID;
  float s1 = 0.f, s2 = 0.f;
  #pragma unroll
  for (int cc = 0; cc < HID; cc++) { float v = xp[cc]; s1 += v * as[cc]; s2 += v * ad[cc]; }
  a_src[t] = s1; a_dst[t] = s2;
}

// ============================ edge pass A1: segment max ================
__global__ void edge_max1(const int* __restrict__ ei, const float* __restrict__ a_src,
                          const float* __restrict__ a_dst, unsigned* __restrict__ amax, int E_) {
  int e = blockIdx.x * blockDim.x + threadIdx.x;
  if (e >= E_) return;
  int s = ei[e], d = ei[E_ + e];
  #pragma unroll
  for (int hh = 0; hh < HEADS; hh++) {
    float al = lrelu(a_src[s * HEADS + hh] + a_dst[d * HEADS + hh]);
    atomicMax(&amax[d * HEADS + hh], fenc(al));
  }
}

// ============================ edge pass B1: segment sum of exp =========
__global__ void edge_sum1(const int* __restrict__ ei, const float* __restrict__ a_src,
                          const float* __restrict__ a_dst, const unsigned* __restrict__ amax,
                          float* __restrict__ denom, int E_) {
  int e = blockIdx.x * blockDim.x + threadIdx.x;
  if (e >= E_) return;
  int s = ei[e], d = ei[E_ + e];
  #pragma unroll
  for (int hh = 0; hh < HEADS; hh++) {
    float al = lrelu(a_src[s * HEADS + hh] + a_dst[d * HEADS + hh]);
    float ex = __expf(al - fdec(amax[d * HEADS + hh]));
    atomicAdd(&denom[d * HEADS + hh], ex);
  }
}

// ============================ edge pass C1: weighted scatter-add =======
__global__ void edge_msg1(const int* __restrict__ ei, const float* __restrict__ a_src,
                          const float* __restrict__ a_dst, const unsigned* __restrict__ amax,
                          const float* __restrict__ denom, const float* __restrict__ xl1,
                          float* __restrict__ out1, int E_) {
  long long t = (long long)blockIdx.x * blockDim.x + threadIdx.x;
  if (t >= (long long)E_ * HEADS) return;
  int e = (int)(t >> 3), hh = (int)(t & 7);
  int s = ei[e], d = ei[E_ + e];
  float al = lrelu(a_src[s * HEADS + hh] + a_dst[d * HEADS + hh]);
  float ex = __expf(al - fdec(amax[d * HEADS + hh]));
  float coef = ex / (denom[d * HEADS + hh] + EPS_GAT);
  const float4* xp = (const float4*)(xl1 + (size_t)s * D1 + hh * HID);
  float4 v0 = xp[0], v1 = xp[1];
  float* op = out1 + (size_t)d * D1 + hh * HID;
  atomicAdd(op + 0, v0.x * coef); atomicAdd(op + 1, v0.y * coef);
  atomicAdd(op + 2, v0.z * coef); atomicAdd(op + 3, v0.w * coef);
  atomicAdd(op + 4, v1.x * coef); atomicAdd(op + 5, v1.y * coef);
  atomicAdd(op + 6, v1.z * coef); atomicAdd(op + 7, v1.w * coef);
}

// ============================ h = elu(out1 + bias1), in place ==========
__global__ void elu_kernel(float* __restrict__ out1, const float* __restrict__ bias1, int n) {
  int i = blockIdx.x * blockDim.x + threadIdx.x;
  if (i >= n) return;
  float v = out1[i] + bias1[i & (D1 - 1)];
  out1[i] = v > 0.f ? v : (__expf(v) - 1.f);
}

// ============================ GEMM2: xl2 = h @ W2  [N,64]x[64,16] ======
// 256 threads = 8 waves; each wave handles one 16-row tile (full 16 cols).
__global__ __launch_bounds__(256) void gemm2_kernel(const float* __restrict__ h,
                                                    const float* __restrict__ W2,
                                                    float* __restrict__ xl2, int nrows) {
  const int wave = threadIdx.x >> 5;
  const int lane = threadIdx.x & 31;
  const int tile = blockIdx.x * 8 + wave;
  const int rowBase = tile * 16;
  if (rowBase >= nrows) return;   // wave-uniform exit
  const int ln = lane & 15;
  const int khalf = (lane >> 4) << 1;
  int arow = rowBase + ln; if (arow >= nrows) arow = nrows - 1;
  const float* ap = h + (size_t)arow * D1;

  v8f c = {};
  #pragma unroll
  for (int k = 0; k < D1; k += 4) {
    v2f a = *(const v2f*)&ap[k + khalf];
    v2f b; b.x = W2[(k + khalf) * NCLS + ln];
           b.y = W2[(k + khalf + 1) * NCLS + ln];
    c = __builtin_amdgcn_wmma_f32_16x16x4_f32(false, a, false, b, (short)0, c,
                                              false, false);
  }
  const int mBase = (lane >> 4) * 8;
  float* outp = xl2 + (size_t)rowBase * NCLS;
  #pragma unroll
  for (int i = 0; i < 8; i++) {
    int row = rowBase + mBase + i;
    if (row < nrows) outp[(size_t)(mBase + i) * NCLS + ln] = c[i];
  }
}

// ============================ attn2: per node (1 head, 16 ch) ==========
__global__ void attn2_kernel(const float* __restrict__ xl2,
                             const float* __restrict__ att_src,
                             const float* __restrict__ att_dst,
                             float* __restrict__ a_src, float* __restrict__ a_dst, int n) {
  int t = blockIdx.x * blockDim.x + threadIdx.x;
  if (t >= n) return;
  const float* xp = xl2 + (size_t)t * NCLS;
  float s1 = 0.f, s2 = 0.f;
  #pragma unroll
  for (int cc = 0; cc < NCLS; cc++) { float v = xp[cc]; s1 += v * att_src[cc]; s2 += v * att_dst[cc]; }
  a_src[t] = s1; a_dst[t] = s2;
}

__global__ void edge_max2(const int* __restrict__ ei, const float* __restrict__ a_src,
                          const float* __restrict__ a_dst, unsigned* __restrict__ amax, int E_) {
  int e = blockIdx.x * blockDim.x + threadIdx.x;
  if (e >= E_) return;
  int s = ei[e], d = ei[E_ + e];
  atomicMax(&amax[d], fenc(lrelu(a_src[s] + a_dst[d])));
}

__global__ void edge_sum2(const int* __restrict__ ei, const float* __restrict__ a_src,
                          const float* __restrict__ a_dst, const unsigned* __restrict__ amax,
                          float* __restrict__ denom, int E_) {
  int e = blockIdx.x * blockDim.x + threadIdx.x;
  if (e >= E_) return;
  int s = ei[e], d = ei[E_ + e];
  float al = lrelu(a_src[s] + a_dst[d]);
  atomicAdd(&denom[d], __expf(al - fdec(amax[d])));
}

__global__ void edge_msg2(const int* __restrict__ ei, const float* __restrict__ a_src,
                          const float* __restrict__ a_dst, const unsigned* __restrict__ amax,
                          const float* __restrict__ denom, const float* __restrict__ xl2,
                          float* __restrict__ out2, int E_) {
  int e = blockIdx.x * blockDim.x + threadIdx.x;
  if (e >= E_) return;
  int s = ei[e], d = ei[E_ + e];
  float al = lrelu(a_src[s] + a_dst[d]);
  float ex = __expf(al - fdec(amax[d]));
  float coef = ex / (denom[d] + EPS_GAT);
  const float4* xp = (const float4*)(xl2 + (size_t)s * NCLS);
  float* op = out2 + (size_t)d * NCLS;
  #pragma unroll
  for (int q = 0; q < 4; q++) {
    float4 v = xp[q];
    atomicAdd(op + q * 4 + 0, v.x * coef); atomicAdd(op + q * 4 + 1, v.y * coef);
    atomicAdd(op + q * 4 + 2, v.z * coef); atomicAdd(op + q * 4 + 3, v.w * coef);
  }
}

__global__ void final_kernel(const float* __restrict__ out2, const float* __restrict__ bias2,
                             float* __restrict__ out, int n) {
  int i = blockIdx.x * blockDim.x + threadIdx.x;
  if (i >= n) return;
  out[i] = out2[i] + bias2[i & (NCLS - 1)];
}

// ============================ launcher ============================
extern "C" void kernel_launch(void* const* d_in, const int* in_sizes, int n_in,
                              void* d_out, int out_size, void* d_ws, size_t ws_size,
                              hipStream_t stream) {
  const float* x        = (const float*)d_in[0];
  const int*   ei       = (const int*)d_in[1];
  const float* W1       = (const float*)d_in[2];
  const float* att_src1 = (const float*)d_in[3];
  const float* att_dst1 = (const float*)d_in[4];
  const float* bias1    = (const float*)d_in[5];
  const float* W2       = (const float*)d_in[6];
  const float* att_src2 = (const float*)d_in[7];
  const float* att_dst2 = (const float*)d_in[8];
  const float* bias2    = (const float*)d_in[9];
  float* out = (float*)d_out;

  const int N = in_sizes[0] / F_IN;
  const int E = in_sizes[1] / 2;

  // workspace layout (floats)
  float* ws = (float*)d_ws;
  size_t o = 0;
  float*    xl1    = ws + o; o += (size_t)N * D1;
  float*    out1   = ws + o; o += (size_t)N * D1;     // becomes h after elu
  float*    a_src1 = ws + o; o += (size_t)N * HEADS;
  float*    a_dst1 = ws + o; o += (size_t)N * HEADS;
  unsigned* amax1  = (unsigned*)(ws + o); o += (size_t)N * HEADS;
  float*    denom1 = ws + o; o += (size_t)N * HEADS;
  float*    xl2    = ws + o; o += (size_t)N * NCLS;
  float*    a_src2 = ws + o; o += (size_t)N;
  float*    a_dst2 = ws + o; o += (size_t)N;
  unsigned* amax2  = (unsigned*)(ws + o); o += (size_t)N;
  float*    denom2 = ws + o; o += (size_t)N;
  float*    out2   = ws + o; o += (size_t)N * NCLS;

  const int TB = 256;
  auto nb = [](long long n, int b) { return (unsigned)((n + b - 1) / b); };

  // init accumulators
  fill_f32<<<nb((long long)N * D1, TB), TB, 0, stream>>>(out1, 0.f, N * D1);
  fill_f32<<<nb((long long)N * HEADS, TB), TB, 0, stream>>>(denom1, 0.f, N * HEADS);
  fill_u32<<<nb((long long)N * HEADS, TB), TB, 0, stream>>>(amax1, ENC_NEG_INF, N * HEADS);
  fill_f32<<<nb((long long)N * NCLS, TB), TB, 0, stream>>>(out2, 0.f, N * NCLS);
  fill_f32<<<nb((long long)N, TB), TB, 0, stream>>>(denom2, 0.f, N);
  fill_u32<<<nb((long long)N, TB), TB, 0, stream>>>(amax2, ENC_NEG_INF, N);

  // ---- layer 1 ----
  gemm1_kernel<<<nb((long long)N, 16), 128, 0, stream>>>(x, W1, xl1, N);
  attn1_kernel<<<nb((long long)N * HEADS, TB), TB, 0, stream>>>(xl1, att_src1, att_dst1,
                                                                a_src1, a_dst1, N);
  edge_max1<<<nb(E, TB), TB, 0, stream>>>(ei, a_src1, a_dst1, amax1, E);
  edge_sum1<<<nb(E, TB), TB, 0, stream>>>(ei, a_src1, a_dst1, amax1, denom1, E);
  edge_msg1<<<nb((long long)E * HEADS, TB), TB, 0, stream>>>(ei, a_src1, a_dst1, amax1,
                                                             denom1, xl1, out1, E);
  elu_kernel<<<nb((long long)N * D1, TB), TB, 0, stream>>>(out1, bias1, N * D1);

  // ---- layer 2 ----
  gemm2_kernel<<<nb(((long long)N + 15) / 16, 8), 256, 0, stream>>>(out1, W2, xl2, N);
  attn2_kernel<<<nb(N, TB), TB, 0, stream>>>(xl2, att_src2, att_dst2, a_src2, a_dst2, N);
  edge_max2<<<nb(E, TB), TB, 0, stream>>>(ei, a_src2, a_dst2, amax2, E);
  edge_sum2<<<nb(E, TB), TB, 0, stream>>>(ei, a_src2, a_dst2, amax2, denom2, E);
  edge_msg2<<<nb(E, TB), TB, 0, stream>>>(ei, a_src2, a_dst2, amax2, denom2, xl2, out2, E);
  final_kernel<<<nb((long long)N * NCLS, TB), TB, 0, stream>>>(out2, bias2, out, N * NCLS);
}